// CompGraphConv_78726750536192
// MI455X (gfx1250) — compile-verified
//
#include <hip/hip_runtime.h>

#define HID 128
#define NNODES 100000
#define NEDGES 500000

typedef __attribute__((ext_vector_type(2))) float v2f;
typedef __attribute__((ext_vector_type(8))) float v8f;

// ---------------------------------------------------------------------------
// Transpose one 128x128 weight so B fragments (two adjacent-K values of one
// output column) become a single contiguous b64 load.
// ---------------------------------------------------------------------------
__global__ __launch_bounds__(256) void transpose_w(const float* __restrict__ W,
                                                   float* __restrict__ WT) {
  int idx = blockIdx.x * blockDim.x + threadIdx.x;
  if (idx >= HID * HID) return;
  int k = idx >> 7;
  int n = idx & 127;
  WT[n * HID + k] = W[k * HID + n];
}

// ---------------------------------------------------------------------------
// Edge scatter: m = n_feat[s] - e_h, atomically segment-summed into buckets.
// 32 lanes per edge, float4 per lane (32*4 = 128 features).
// ---------------------------------------------------------------------------
__global__ __launch_bounds__(256) void edge_scatter(
    const float* __restrict__ n_feat, const float* __restrict__ e_rel,
    const float* __restrict__ e_inv, const int* __restrict__ src,
    const int* __restrict__ dst, float* __restrict__ comp_rel,
    float* __restrict__ comp_inv, float* __restrict__ deg_rel,
    float* __restrict__ deg_inv, int n_edges) {
  int tid = blockIdx.x * blockDim.x + threadIdx.x;
  int e = tid >> 5;
  int lane = tid & 31;
  if (e >= n_edges) return;
  int s = src[e];
  int d = dst[e];
  const float4* nfs = (const float4*)(n_feat + (size_t)s * HID);
  const float4* nfd = (const float4*)(n_feat + (size_t)d * HID);
  const float4* er = (const float4*)(e_rel + (size_t)e * HID);
  const float4* ei = (const float4*)(e_inv + (size_t)e * HID);
  float4 a = nfs[lane], b = er[lane];
  float4 c = nfd[lane], g = ei[lane];
  float4 m1 = make_float4(a.x - b.x, a.y - b.y, a.z - b.z, a.w - b.w);
  float4 m2 = make_float4(c.x - g.x, c.y - g.y, c.z - g.z, c.w - g.w);
  float* cr = comp_rel + (size_t)d * HID + lane * 4;
  float* ci = comp_inv + (size_t)s * HID + lane * 4;
  atomicAdd(cr + 0, m1.x);
  atomicAdd(cr + 1, m1.y);
  atomicAdd(cr + 2, m1.z);
  atomicAdd(cr + 3, m1.w);
  atomicAdd(ci + 0, m2.x);
  atomicAdd(ci + 1, m2.y);
  atomicAdd(ci + 2, m2.z);
  atomicAdd(ci + 3, m2.w);
  if (lane == 0) {
    atomicAdd(deg_rel + d, 1.0f);
    atomicAdd(deg_inv + s, 1.0f);
  }
}

// ---------------------------------------------------------------------------
// Edge GEMM: Y = relu(X @ W + bias).  256 threads = 8 waves, each wave owns a
// 16-row block (128 rows/block).  B is staged once per block into LDS in
// *fragment order* ldsB[kstep][nt][lane]:
//   - hot-loop B read = ds_load_b64 of 32 consecutive float2s (256B
//     contiguous -> every LDS bank hit exactly once, conflict-free)
//   - 16-bit DS offset covers all of it (kt*2048 + nt*256 <= 65280)
// This cuts per-wave global B traffic 8x and moves it off the VMEM port so
// the A-stream loads and WMMAs pipeline cleanly.
// ---------------------------------------------------------------------------
__global__ __launch_bounds__(256) void edge_gemm_relu(
    const float* __restrict__ X, const float* __restrict__ WT,
    const float* __restrict__ bias, float* __restrict__ Y, int rows) {
  __shared__ v2f ldsB[8192];  // 64 KB: 32 ksteps x 8 ntiles x 32 lanes

  const int lane = threadIdx.x & 31;
  const int wave = threadIdx.x >> 5;
  const int half = lane >> 4;  // 0: K lo pair, 1: K hi pair
  const int t = lane & 15;     // A row / B,C column within tile

  // Cooperative fill: entry i holds the pair lane (i&31) needs at kstep i>>8,
  // ntile (i>>5)&7.  Pair value = WT[(nt*16+t)*64 + 2*kt + half].
  {
    const v2f* __restrict__ Wp = (const v2f*)WT;
#pragma unroll
    for (int j = 0; j < 32; ++j) {
      int i = j * 256 + threadIdx.x;
      int kt = i >> 8;
      int nt = (i >> 5) & 7;
      int ln = i & 31;
      ldsB[i] = Wp[((nt * 16 + (ln & 15)) << 6) + 2 * kt + (ln >> 4)];
    }
  }
  __syncthreads();

  const int rowBlock = blockIdx.x * 8 + wave;
  const int row0 = rowBlock * 16;
  if (row0 >= rows) return;  // wave-uniform: EXEC stays all-ones for WMMA

  const v2f* __restrict__ Arow =
      (const v2f*)(X + (size_t)row0 * HID) + t * 64 + half;
  const v2f* __restrict__ Bl = ldsB + lane;

  float bcol[8];
#pragma unroll
  for (int nt = 0; nt < 8; ++nt) bcol[nt] = bias[nt * 16 + t];

  v8f acc[8] = {};
#pragma unroll 4
  for (int kt = 0; kt < 32; ++kt) {  // K advances by 4 per step
    v2f a = Arow[2 * kt];
#pragma unroll
    for (int nt = 0; nt < 8; ++nt) {
      v2f b = Bl[kt * 256 + nt * 32];
      acc[nt] = __builtin_amdgcn_wmma_f32_16x16x4_f32(
          false, a, false, b, (short)0, acc[nt], false, false);
    }
  }

#pragma unroll
  for (int nt = 0; nt < 8; ++nt) {
#pragma unroll
    for (int v = 0; v < 8; ++v) {
      float val = acc[nt][v] + bcol[nt];
      val = fmaxf(val, 0.0f);
      Y[(size_t)(row0 + v + 8 * half) * HID + nt * 16 + t] = val;
    }
  }
}

// ---------------------------------------------------------------------------
// Fused node update:
// n_out = relu( 2/3*(norm_r*comp_rel @ W_I + norm_i*comp_inv @ W_O)
//             + 1/3*(n_feat @ W_S) + combined_bias )
// Scales folded into A fragments; bias combined in the epilogue.  Three B
// matrices (192 KB) -> keep them on the global path (L2-resident).
// ---------------------------------------------------------------------------
__global__ __launch_bounds__(128) void node_fused_gemm(
    const float* __restrict__ n_feat, const float* __restrict__ comp_rel,
    const float* __restrict__ comp_inv, const float* __restrict__ deg_rel,
    const float* __restrict__ deg_inv, const float* __restrict__ WT_I,
    const float* __restrict__ WT_O, const float* __restrict__ WT_S,
    const float* __restrict__ b_I, const float* __restrict__ b_O,
    const float* __restrict__ b_S, float* __restrict__ n_out, int n_nodes) {
  const int lane = threadIdx.x & 31;
  const int wave = threadIdx.x >> 5;
  const int half = lane >> 4;
  const int t = lane & 15;
  const int rowBlock = blockIdx.x * 4 + wave;
  const int row0 = rowBlock * 16;
  if (row0 >= n_nodes) return;

  const float TWO3 = 2.0f / 3.0f;
  const float ONE3 = 1.0f / 3.0f;
  const int row = row0 + t;  // the A-matrix row this lane owns
  const float sR = TWO3 * rsqrtf(fmaxf(deg_rel[row], 1.0f));
  const float sI = TWO3 * rsqrtf(fmaxf(deg_inv[row], 1.0f));

  const v2f* __restrict__ A1 =
      (const v2f*)(comp_rel + (size_t)row0 * HID) + t * 64 + half;
  const v2f* __restrict__ A2 =
      (const v2f*)(comp_inv + (size_t)row0 * HID) + t * 64 + half;
  const v2f* __restrict__ A3 =
      (const v2f*)(n_feat + (size_t)row0 * HID) + t * 64 + half;
  const v2f* __restrict__ B1 = (const v2f*)WT_I + t * 64 + half;
  const v2f* __restrict__ B2 = (const v2f*)WT_O + t * 64 + half;
  const v2f* __restrict__ B3 = (const v2f*)WT_S + t * 64 + half;

  float bcol[8];
#pragma unroll
  for (int nt = 0; nt < 8; ++nt) {
    int n = nt * 16 + t;
    bcol[nt] = TWO3 * (b_I[n] + b_O[n]) + ONE3 * b_S[n];
  }

  v8f acc[8] = {};
#pragma unroll 2
  for (int k2 = 0; k2 < 64; k2 += 2) {
    v2f a1 = A1[k2];
    a1.x *= sR;
    a1.y *= sR;
    v2f a2 = A2[k2];
    a2.x *= sI;
    a2.y *= sI;
    v2f a3 = A3[k2];
    a3.x *= ONE3;
    a3.y *= ONE3;
#pragma unroll
    for (int nt = 0; nt < 8; ++nt) {
      int off = k2 + nt * 1024;
      acc[nt] = __builtin_amdgcn_wmma_f32_16x16x4_f32(
          false, a1, false, B1[off], (short)0, acc[nt], false, false);
      acc[nt] = __builtin_amdgcn_wmma_f32_16x16x4_f32(
          false, a2, false, B2[off], (short)0, acc[nt], false, false);
      acc[nt] = __builtin_amdgcn_wmma_f32_16x16x4_f32(
          false, a3, false, B3[off], (short)0, acc[nt], false, false);
    }
  }

#pragma unroll
  for (int nt = 0; nt < 8; ++nt) {
#pragma unroll
    for (int v = 0; v < 8; ++v) {
      float val = acc[nt][v] + bcol[nt];
      val = fmaxf(val, 0.0f);
      n_out[(size_t)(row0 + v + 8 * half) * HID + nt * 16 + t] = val;
    }
  }
}

extern "C" void kernel_launch(void* const* d_in, const int* in_sizes, int n_in,
                              void* d_out, int out_size, void* d_ws,
                              size_t ws_size, hipStream_t stream) {
  (void)in_sizes;
  (void)n_in;
  (void)out_size;
  (void)ws_size;
  const float* n_feat = (const float*)d_in[0];
  const float* e_rel = (const float*)d_in[1];
  const float* e_inv = (const float*)d_in[2];
  const int* src = (const int*)d_in[3];
  const int* dst = (const int*)d_in[4];
  const float* W_I = (const float*)d_in[5];
  const float* W_O = (const float*)d_in[6];
  const float* W_S = (const float*)d_in[7];
  const float* W_rel = (const float*)d_in[8];
  const float* b_I = (const float*)d_in[9];
  const float* b_O = (const float*)d_in[10];
  const float* b_S = (const float*)d_in[11];
  const float* b_rel = (const float*)d_in[12];

  // Workspace layout (floats): buckets, degrees, transposed weights.
  float* ws = (float*)d_ws;
  float* comp_rel = ws;
  float* comp_inv = comp_rel + (size_t)NNODES * HID;
  float* deg_rel = comp_inv + (size_t)NNODES * HID;
  float* deg_inv = deg_rel + NNODES;
  float* WT_I = deg_inv + NNODES;
  float* WT_O = WT_I + HID * HID;
  float* WT_S = WT_O + HID * HID;
  float* WT_rel = WT_S + HID * HID;

  float* n_out = (float*)d_out;
  float* e_out_rel = n_out + (size_t)NNODES * HID;
  float* e_out_inv = e_out_rel + (size_t)NEDGES * HID;

  // Zero segment-sum buckets + degree counters (capturable stream memset).
  hipMemsetAsync(comp_rel, 0,
                 ((size_t)2 * NNODES * HID + 2 * NNODES) * sizeof(float),
                 stream);

  transpose_w<<<64, 256, 0, stream>>>(W_I, WT_I);
  transpose_w<<<64, 256, 0, stream>>>(W_O, WT_O);
  transpose_w<<<64, 256, 0, stream>>>(W_S, WT_S);
  transpose_w<<<64, 256, 0, stream>>>(W_rel, WT_rel);

  // 32 lanes per edge -> NEDGES*32 threads.
  edge_scatter<<<(NEDGES * 32) / 256, 256, 0, stream>>>(
      n_feat, e_rel, e_inv, src, dst, comp_rel, comp_inv, deg_rel, deg_inv,
      NEDGES);

  // 500000 rows / 16 per wave / 8 waves per block = 3906.25 -> 3907 blocks.
  edge_gemm_relu<<<3907, 256, 0, stream>>>(e_rel, WT_rel, b_rel, e_out_rel,
                                           NEDGES);
  edge_gemm_relu<<<3907, 256, 0, stream>>>(e_inv, WT_rel, b_rel, e_out_inv,
                                           NEDGES);

  // 100000 rows / 16 / 4 = 1562.5 -> 1563 blocks.
  node_fused_gemm<<<1563, 128, 0, stream>>>(n_feat, comp_rel, comp_inv,
                                            deg_rel, deg_inv, WT_I, WT_O, WT_S,
                                            b_I, b_O, b_S, n_out, NNODES);
}